// Fold_45174466019398
// MI455X (gfx1250) — compile-verified
//
#include <hip/hip_runtime.h>

// Fold (col2im) 3D: x (1,16,8,8,8,4096) f32 -> out (1,16,68,68,68) f32
//   KS=(8,8,8), ST=(4,4,4), n=16 windows/dim, out dim = 4*15+8 = 68.
// Each input element maps to exactly one output cell; each output cell sums
// <= 8 inputs. Memory-bound: ~147 MiB total -> ~6.3us at 23.3 TB/s.
// Strategy: per-block async-DMA a 32 KiB input tile into LDS (coalesced,
// latency hidden on ASYNCcnt), do the irregular 8-way gather from LDS,
// write each output exactly once (no atomics, no pre-zero needed).

#define KSZ   8
#define NWIN  16
#define OUTD  68
#define CH    16
#define LWIN  4096   // 16^3
#define NPATCH 512   // 8^3

__global__ __launch_bounds__(256)
void fold3d_kernel(const float* __restrict__ x, float* __restrict__ out) {
    // tile[k1][k2][k3][n3] : 8*8*8*16 floats = 32 KiB
    __shared__ float tile[NPATCH * 16];

    const int tid = threadIdx.x;
    const int N2  = blockIdx.x;   // 0..16
    const int N1  = blockIdx.y;   // 0..16
    const int c   = blockIdx.z;   // 0..15

    const unsigned long long xbase =
        (unsigned long long)(x + (size_t)c * NPATCH * LWIN);

    // ---------------- Stage: global -> LDS via async DMA ----------------
    // 2048 segments of 16B: segment s -> row = s>>2 (k1,k2,k3), part = s&3.
    // LDS dest elem = row*16 + part*4 = s*4.
#pragma unroll
    for (int i = 0; i < 8; ++i) {
        const int s    = tid + (i << 8);       // 0..2047
        const int row  = s >> 2;               // 0..511
        const int part = s & 3;                // 0..3
        const int k1 = row >> 6;
        const int k2 = (row >> 3) & 7;
        const int n1 = N1 - (k1 >> 2);         // k1>=4 -> window N1-1
        const int n2 = N2 - (k2 >> 2);
        const bool valid = (n1 >= 0) && (n1 <= 15) && (n2 >= 0) && (n2 <= 15);
        if (valid) {
            const int goff = (row * LWIN + n1 * 256 + n2 * 16 + part * 4) * 4; // bytes
            const unsigned lds_off =
                (unsigned)(unsigned long long)(&tile[s * 4]);  // low 32b = LDS byte offset
            asm volatile("global_load_async_to_lds_b128 %0, %1, %2"
                         :
                         : "v"(lds_off), "v"(goff), "s"(xbase)
                         : "memory");
        }
    }
    asm volatile("s_wait_asynccnt 0" ::: "memory");
    __syncthreads();

    // ---------------- Gather: 8 masked LDS reads per output -------------
    const int p1b = N1 * 4;
    const int p2b = N2 * 4;
    const bool ok1[2] = { N1 <= 15, N1 >= 1 };
    const bool ok2[2] = { N2 <= 15, N2 >= 1 };

    for (int o = tid; o < 4 * 4 * OUTD; o += 256) {
        const int j1  = o / (4 * OUTD);
        const int rem = o - j1 * (4 * OUTD);
        const int j2  = rem / OUTD;
        const int p3  = rem - j2 * OUTD;

        const int m3 = p3 >> 2;
        const int r3 = p3 & 3;
        const bool ok3[2] = { m3 <= 15, m3 >= 1 };

        float acc = 0.0f;
#pragma unroll
        for (int a1 = 0; a1 < 2; ++a1) {
#pragma unroll
            for (int a2 = 0; a2 < 2; ++a2) {
#pragma unroll
                for (int a3 = 0; a3 < 2; ++a3) {
                    if (ok1[a1] && ok2[a2] && ok3[a3]) {
                        const int k1 = j1 + 4 * a1;
                        const int k2 = j2 + 4 * a2;
                        const int k3 = r3 + 4 * a3;
                        const int n3 = m3 - a3;
                        acc += tile[((k1 * 8 + k2) * 8 + k3) * 16 + n3];
                    }
                }
            }
        }

        const size_t oidx =
            (((size_t)c * OUTD + (p1b + j1)) * OUTD + (p2b + j2)) * OUTD + p3;
        __builtin_nontemporal_store(acc, &out[oidx]);
    }
}

extern "C" void kernel_launch(void* const* d_in, const int* in_sizes, int n_in,
                              void* d_out, int out_size, void* d_ws, size_t ws_size,
                              hipStream_t stream) {
    const float* x  = (const float*)d_in[0];
    float* out      = (float*)d_out;
    dim3 grid(17, 17, 16);   // (N2, N1, c)
    fold3d_kernel<<<grid, 256, 0, stream>>>(x, out);
}